// WavetableModel_66649302499810
// MI455X (gfx1250) — compile-verified
//
#include <hip/hip_runtime.h>

typedef float v2f  __attribute__((ext_vector_type(2)));
typedef float v8f  __attribute__((ext_vector_type(8)));
typedef unsigned int u32x4 __attribute__((ext_vector_type(4)));
typedef int  i32x4 __attribute__((ext_vector_type(4)));
typedef int  i32x8 __attribute__((ext_vector_type(8)));

#define TABLE_N   4096
#define NSAMP     32768
#define THREADS   256
#define SPB       128   // samples per block (8 waves * 16)

// sigma = STD + WIN_EPS = 0.01 + 1e-8
// log_norm = -ln(sigma) - 0.5*ln(2*pi) = 3.68623065...
__global__ __launch_bounds__(THREADS)
void wavetable_wmma_kernel(const float* __restrict__ f0p,
                           const float* __restrict__ table,
                           float* __restrict__ out)
{
    __shared__ float lut[TABLE_N];

    const int tid = threadIdx.x;

    // ---- Stage the 16KB table into LDS via the Tensor Data Mover ----
    // D# group 0: count=1 (valid), lds_addr, 57-bit global_addr, type=2.
    // D# group 1: data_size=4B (code 2), tensor_dim0=4096, tensor_dim1=1,
    //             tile_dim0=4096, tile_dim1=1, tensor_dim0_stride=4096.
    // Groups 2/3 zero (dims beyond 2D unused).
    if (tid < 32) {  // one wave issues the DMA (EXEC-independent, SGPR descriptor)
        const unsigned long long gaddr = (unsigned long long)(const void*)table;
        const unsigned int lds_addr = (unsigned int)(unsigned long long)(const void*)lut;

        u32x4 g0;
        g0.x = 1u;                                   // count=1, user mode
        g0.y = lds_addr;                             // lds_addr [63:32]
        g0.z = (unsigned int)(gaddr & 0xFFFFFFFFull);          // global_addr lo
        g0.w = (unsigned int)((gaddr >> 32) & 0x01FFFFFFull)   // global_addr hi
               | 0x80000000u;                        // type=2 ("image")

        i32x8 g1;
        g1[0] = 0x00020000;                          // wg_mask=0, data_size=2 (4B)
        g1[1] = (int)(0x1000u << 16);                // tensor_dim0 lo16 = 4096
        g1[2] = 0x00010000;                          // tensor_dim0 hi=0, tensor_dim1=1
        g1[3] = (int)(0x1000u << 16);                // tensor_dim1 hi=0, tile_dim0=4096
        g1[4] = 0x00000001;                          // tile_dim1=1, tile_dim2=0
        g1[5] = 0x00001000;                          // tensor_dim0_stride = 4096
        g1[6] = 0;
        g1[7] = 0;

        i32x4 gz4 = {0, 0, 0, 0};
        i32x8 gz8 = {0, 0, 0, 0, 0, 0, 0, 0};
        __builtin_amdgcn_tensor_load_to_lds(g0, g1, gz4, gz4, gz8, 0); // TENSORcnt++
        __builtin_amdgcn_s_wait_tensorcnt(0);                          // s_wait_tensorcnt 0
    }
    __syncthreads();

    // Idempotent fallback staging (rewrites identical values): guarantees the
    // LUT contents regardless of TDM descriptor interpretation; one-time 16KB.
    #pragma unroll
    for (int i = 0; i < TABLE_N / THREADS; ++i) {
        const int idx = tid + i * THREADS;
        lut[idx] = table[idx];
    }
    __syncthreads();

    const float f0a      = fabsf(f0p[0]);
    const float sigma    = 0.01000001f;      // STD + WIN_EPS (fp32)
    const float inv_sig  = 1.0f / sigma;
    const float log_norm = 3.68623065f;      // -ln(sigma) - 0.5*ln(2*pi)
    const float inv_step = 1.0f / 4095.0f;   // linspace(0,1,4096) spacing

    const int lane = tid & 31;
    const int wave = tid >> 5;
    const int base = blockIdx.x * SPB + wave * 16;   // first sample of this wave

    // A-matrix layout (f32 16x4): lanes 0-15 -> M=lane, K = kb+{0,1};
    //                             lanes 16-31 -> M=lane-16, K = kb+{2,3}.
    const int   m    = lane & 15;
    const int   koff = (lane < 16) ? 0 : 2;
    const float mu   = fmodf((float)(base + m + 1) * f0a, 1.0f);

    v8f c0 = {};
    v8f c1 = {};

    for (int kb = 0; kb < TABLE_N; kb += 8) {
        {
            const int   k0 = kb + koff;
            const float z0 = ((float)k0       * inv_step - mu) * inv_sig;
            const float z1 = ((float)(k0 + 1) * inv_step - mu) * inv_sig;
            v2f a, b;
            a.x = __expf(fmaf(-0.5f * z0, z0, log_norm));
            a.y = __expf(fmaf(-0.5f * z1, z1, log_norm));
            b.x = lut[k0];
            b.y = lut[k0 + 1];
            c0 = __builtin_amdgcn_wmma_f32_16x16x4_f32(
                     false, a, false, b, (short)0, c0, false, false);
        }
        {
            const int   k0 = kb + 4 + koff;
            const float z0 = ((float)k0       * inv_step - mu) * inv_sig;
            const float z1 = ((float)(k0 + 1) * inv_step - mu) * inv_sig;
            v2f a, b;
            a.x = __expf(fmaf(-0.5f * z0, z0, log_norm));
            a.y = __expf(fmaf(-0.5f * z1, z1, log_norm));
            b.x = lut[k0];
            b.y = lut[k0 + 1];
            c1 = __builtin_amdgcn_wmma_f32_16x16x4_f32(
                     false, a, false, b, (short)0, c1, false, false);
        }
    }

    const v8f c = c0 + c1;

    // D layout: lanes 0-15 hold M = v (VGPR v), lanes 16-31 hold M = v+8.
    // Every N column is identical (B was broadcast across N), so lane j (j<8)
    // stores row j, lane 16+j stores row 8+j.
    if ((lane & 15) < 8) {
        const int sm = (lane & 7) + ((lane >= 16) ? 8 : 0);
        const int n  = base + sm;
        // Row max of the Gaussian = value at the nearest grid point to mu.
        const float muv   = fmodf((float)(n + 1) * f0a, 1.0f);
        const float tst   = floorf(muv * 4095.0f + 0.5f);   // nearest grid idx
        const float zs    = (tst * inv_step - muv) * inv_sig;
        const float peak  = __expf(fmaf(-0.5f * zs, zs, log_norm));
        const float scale = 1.0f / (peak + 1.0e-8f);
        out[n] = c[lane & 7] * scale;
    }
}

extern "C" void kernel_launch(void* const* d_in, const int* in_sizes, int n_in,
                              void* d_out, int out_size, void* d_ws, size_t ws_size,
                              hipStream_t stream)
{
    // setup_inputs order: x (unused), f0, table, n_samples
    const float* f0    = (const float*)d_in[1];
    const float* table = (const float*)d_in[2];
    float*       out   = (float*)d_out;

    dim3 grid(NSAMP / SPB);
    dim3 block(THREADS);
    wavetable_wmma_kernel<<<grid, block, 0, stream>>>(f0, table, out);
}